// CBAMSLayer_59545426591961
// MI455X (gfx1250) — compile-verified
//
#include <hip/hip_runtime.h>
#include <hip/hip_bf16.h>
#include <math.h>

#define N_   32
#define C_   256
#define H_   56
#define W_   56
#define HW_  (H_*W_)          // 3136
#define EPS_ 1e-5f
#define CCHUNK 32

typedef __attribute__((ext_vector_type(2))) float v2f;
typedef __attribute__((ext_vector_type(8))) float v8f;

// ---------------------------------------------------------------------------
// Kernel 1: per-(n,hw) channel max + mean over C=256. float4 over hw so each
// c-iteration is a fully coalesced 512B/wave load. ~103 MB read, 0.8 MB write.
// ---------------------------------------------------------------------------
__global__ void pool_kernel(const float* __restrict__ x, float* __restrict__ pool) {
  int tid = blockIdx.x * blockDim.x + threadIdx.x;        // 0 .. N*HW/4-1
  int n   = tid / (HW_ / 4);
  int hw4 = tid - n * (HW_ / 4);
  const float4* xp = (const float4*)(x + (size_t)n * C_ * HW_) + hw4;
  float4 mx = xp[0];
  float4 sm = mx;
#pragma unroll 4
  for (int c = 1; c < C_; ++c) {
    float4 v = xp[c * (HW_ / 4)];
    if ((c & 7) == 1 && (c + 8) < C_)
      __builtin_prefetch((const void*)(xp + (c + 8) * (HW_ / 4)), 0, 1);
    mx.x = fmaxf(mx.x, v.x); mx.y = fmaxf(mx.y, v.y);
    mx.z = fmaxf(mx.z, v.z); mx.w = fmaxf(mx.w, v.w);
    sm.x += v.x; sm.y += v.y; sm.z += v.z; sm.w += v.w;
  }
  const float inv = 1.0f / (float)C_;
  float4 mn = make_float4(sm.x * inv, sm.y * inv, sm.z * inv, sm.w * inv);
  ((float4*)(pool + (size_t)n * 2 * HW_))[hw4]       = mx;  // channel 0: max
  ((float4*)(pool + (size_t)n * 2 * HW_ + HW_))[hw4] = mn;  // channel 1: mean
}

// ---------------------------------------------------------------------------
// Kernel 2: 5x5 conv (2->1 ch, SAME) as im2col GEMM on V_WMMA_F32_16X16X4_F32.
// One wave per 16-pixel tile; K = 50 padded to 52 -> 13 WMMA accumulations.
// A = weights replicated over M rows; B columns = 16 output pixels' patches.
// ---------------------------------------------------------------------------
__device__ __forceinline__ float patch_val(const float* __restrict__ pool,
                                           int n, int r, int q, int k) {
  if (k >= 50) return 0.0f;
  int ch = (k >= 25) ? 1 : 0;
  int t  = k - ch * 25;
  int dy = t / 5 - 2;
  int dx = t - (t / 5) * 5 - 2;
  int rr = r + dy, qq = q + dx;
  if ((unsigned)rr >= (unsigned)H_ || (unsigned)qq >= (unsigned)W_) return 0.0f;
  return pool[((size_t)(n * 2 + ch)) * HW_ + rr * W_ + qq];
}

__global__ void conv_wmma_kernel(const float* __restrict__ pool,
                                 const float* __restrict__ w,
                                 float* __restrict__ y) {
  int gid  = blockIdx.x * blockDim.x + threadIdx.x;
  int wave = gid >> 5;
  int lane = gid & 31;
  int n    = wave / (HW_ / 16);
  int t    = wave - n * (HW_ / 16);
  int pix  = t * 16 + (lane & 15);
  int r    = pix / W_;
  int q    = pix - r * W_;
  // A/B 32-bit fragment layout: lanes 0-15 carry K = k0+0 (reg0), k0+1 (reg1);
  // lanes 16-31 carry K = k0+2 (reg0), k0+3 (reg1).
  int koff = (lane >= 16) ? 2 : 0;
  v8f acc = {};
#pragma unroll
  for (int k0 = 0; k0 < 52; k0 += 4) {
    int ka = k0 + koff;
    v2f a, b;
    a.x = (ka     < 50) ? w[ka]     : 0.0f;   // weights broadcast over M rows
    a.y = (ka + 1 < 50) ? w[ka + 1] : 0.0f;
    b.x = patch_val(pool, n, r, q, ka);       // patch for pixel column
    b.y = patch_val(pool, n, r, q, ka + 1);
    acc = __builtin_amdgcn_wmma_f32_16x16x4_f32(
        /*neg_a=*/false, a, /*neg_b=*/false, b,
        /*c_mod=*/(short)0, acc, /*reuse_a=*/false, /*reuse_b=*/false);
  }
  // D row M=0 (acc[0], lanes 0-15, N=lane) holds the 16 outputs of this tile.
  if (lane < 16) y[(size_t)n * HW_ + pix] = acc[0];
}

// ---------------------------------------------------------------------------
// Kernel 3: BN training stats over 100,352 conv outputs; fold gamma/beta into
// y_bn = y*scale + shift.  Single 1024-thread block + LDS tree reduce.
// ---------------------------------------------------------------------------
__global__ void stats_kernel(const float* __restrict__ y,
                             const float* __restrict__ gamma,
                             const float* __restrict__ beta,
                             float* __restrict__ st) {
  __shared__ float ssum[1024];
  __shared__ float ssq[1024];
  int tid = threadIdx.x;
  float s = 0.0f, sq = 0.0f;
  for (int i = tid; i < N_ * HW_; i += 1024) {
    float v = y[i];
    s += v; sq += v * v;
  }
  ssum[tid] = s; ssq[tid] = sq;
  __syncthreads();
  for (int off = 512; off > 0; off >>= 1) {
    if (tid < off) { ssum[tid] += ssum[tid + off]; ssq[tid] += ssq[tid + off]; }
    __syncthreads();
  }
  if (tid == 0) {
    const float inv_m = 1.0f / (float)(N_ * HW_);
    float mean = ssum[0] * inv_m;
    float var  = ssq[0] * inv_m - mean * mean;
    float rinv = rsqrtf(var + EPS_);
    float scale = gamma[0] * rinv;
    st[0] = scale;
    st[1] = beta[0] - mean * scale;
  }
}

// ---------------------------------------------------------------------------
// Kernel 4: out = x * sigmoid(sigmoid(y_bn)). Gate computed once per float4
// of spatial positions, applied across a 32-channel chunk (coalesced).
// ~103 MB read + ~103 MB write.
// ---------------------------------------------------------------------------
__device__ __forceinline__ float dsig(float z) {
  float s1 = 1.0f / (1.0f + __expf(-z));
  return 1.0f / (1.0f + __expf(-s1));
}

__global__ void gate_kernel(const float* __restrict__ x,
                            const float* __restrict__ y,
                            const float* __restrict__ st,
                            float* __restrict__ out) {
  const int hw4n = HW_ / 4;                    // 784
  const int perN = (C_ / CCHUNK) * hw4n;       // 6272
  int tid = blockIdx.x * blockDim.x + threadIdx.x;
  int n   = tid / perN;
  int rem = tid - n * perN;
  int cc  = rem / hw4n;
  int hw4 = rem - cc * hw4n;
  float scale = st[0], shift = st[1];
  float4 yv = ((const float4*)(y + (size_t)n * HW_))[hw4];
  float4 g;
  g.x = dsig(yv.x * scale + shift);
  g.y = dsig(yv.y * scale + shift);
  g.z = dsig(yv.z * scale + shift);
  g.w = dsig(yv.w * scale + shift);
  int c0 = cc * CCHUNK;
  const float4* xp = (const float4*)(x   + ((size_t)n * C_ + c0) * HW_) + hw4;
  float4*       op = (float4*)      (out + ((size_t)n * C_ + c0) * HW_) + hw4;
#pragma unroll 4
  for (int c = 0; c < CCHUNK; ++c) {
    float4 v = xp[c * hw4n];
    v.x *= g.x; v.y *= g.y; v.z *= g.z; v.w *= g.w;
    op[c * hw4n] = v;
  }
}

// ---------------------------------------------------------------------------
extern "C" void kernel_launch(void* const* d_in, const int* in_sizes, int n_in,
                              void* d_out, int out_size, void* d_ws, size_t ws_size,
                              hipStream_t stream) {
  const float* x     = (const float*)d_in[0];   // [32,256,56,56]
  const float* cw    = (const float*)d_in[1];   // [1,2,5,5] = 50 floats
  const float* gamma = (const float*)d_in[2];   // [1]
  const float* beta  = (const float*)d_in[3];   // [1]
  float* out = (float*)d_out;

  float* pool  = (float*)d_ws;                  // [32][2][3136]  (802,816 B)
  float* yconv = pool + (size_t)N_ * 2 * HW_;   // [32][3136]     (401,408 B)
  float* st    = yconv + (size_t)N_ * HW_;      // [2]

  // 1) channel max/mean pooling: 25,088 threads
  pool_kernel<<<(N_ * (HW_ / 4)) / 256, 256, 0, stream>>>(x, pool);
  // 2) 5x5 conv via fp32 WMMA: 6,272 waves (one per 16-pixel tile)
  conv_wmma_kernel<<<(N_ * (HW_ / 16) * 32) / 256, 256, 0, stream>>>(pool, cw, yconv);
  // 3) BN batch statistics -> scale/shift
  stats_kernel<<<1, 1024, 0, stream>>>(yconv, gamma, beta, st);
  // 4) gating pass: 200,704 threads, 32 channels each
  gate_kernel<<<(N_ * (C_ / CCHUNK) * (HW_ / 4)) / 256, 256, 0, stream>>>(x, yconv, st, out);
}